// Model_28793460752886
// MI455X (gfx1250) — compile-verified
//
#include <hip/hip_runtime.h>
#include <math.h>

// ---------------------------------------------------------------------------
// BiLSTM-CRF (B=128, T=512, D=H=100, 4H=400, L=25) for gfx1250 (MI455X).
// fp32 end-to-end using V_WMMA_F32_16X16X4_F32 for all GEMM work.
// ---------------------------------------------------------------------------

typedef __attribute__((ext_vector_type(2))) float v2f;
typedef __attribute__((ext_vector_type(8))) float v8f;

static constexpr int kB  = 128;    // batch
static constexpr int kT  = 512;    // sequence length
static constexpr int kD  = 100;    // embedding dim
static constexpr int kH  = 100;    // hidden per direction
static constexpr int kG  = 400;    // 4*H gates
static constexpr int kL  = 25;     // CRF labels
static constexpr int kM  = kB * kT;        // 65536 GEMM rows
static constexpr int kMT = kM / 16;        // 4096 M-tiles
static constexpr int kNT_G = kG / 16;      // 25 N-tiles for gates

__device__ __forceinline__ v8f wmma_f32_4(v2f a, v2f b, v8f c) {
  // D = A(16x4,f32) * B(4x16,f32) + C(16x16,f32)
  return __builtin_amdgcn_wmma_f32_16x16x4_f32(false, a, false, b,
                                               (short)0, c, false, false);
}

__device__ __forceinline__ float sigmoidf(float x) {
  return 1.0f / (1.0f + expf(-x));
}

// ---------------------------------------------------------------------------
// Kernel 1: xg[t*B+b, g] = emb[tok[b,t]] . w_ih[g,:] + b_ih[g] + b_hh[g]
// Embedding gather fused into the WMMA A-tile load.
// One wave per 16x16 output tile. grid = 12800 blocks x 256 thr (8 waves).
// ---------------------------------------------------------------------------
__global__ __launch_bounds__(256) void k_input_proj(
    const int* __restrict__ tok,      // [B,T]
    const float* __restrict__ emb,    // [V,D]
    const float* __restrict__ w_ih,   // [4H,D]
    const float* __restrict__ b_ih,   // [4H]
    const float* __restrict__ b_hh,   // [4H]
    float* __restrict__ xg)           // [T*B, 4H]
{
  const int lane = threadIdx.x & 31;
  const int wave = threadIdx.x >> 5;
  const int gw   = blockIdx.x * (blockDim.x >> 5) + wave;
  if (gw >= kMT * kNT_G) return;

  const int tile_m = gw / kNT_G;
  const int tile_n = gw % kNT_G;
  const int m0 = tile_m * 16;
  const int n0 = tile_n * 16;

  const int half = lane >> 4;        // 0: K even pair, 1: K odd pair (+2)
  const int lq   = lane & 15;

  // A row: m = t*B + b  ->  token = tok[b*T + t]
  const int m = m0 + lq;
  const int t = m >> 7;              // m / 128
  const int b = m & 127;             // m % 128
  const float* __restrict__ arow = emb + (size_t)tok[b * kT + t] * kD;

  const int n = n0 + lq;             // B column (gate index)
  const float* __restrict__ brow = w_ih + (size_t)n * kD; // B[k][n] = w_ih[n][k]
  const float bias = b_ih[n] + b_hh[n];

  v8f acc = {};
  for (int k0 = 0; k0 < kD; k0 += 4) {
    const int ka = k0 + half * 2;
    v2f av; av.x = arow[ka]; av.y = arow[ka + 1];
    v2f bv; bv.x = brow[ka]; bv.y = brow[ka + 1];
    acc = wmma_f32_4(av, bv, acc);
  }

  // D layout: VGPR r -> rows m0+r (lanes 0-15) / m0+r+8 (lanes 16-31), col n0+lq
  const int col = n0 + lq;
#pragma unroll
  for (int r = 0; r < 8; ++r) {
    const int row = m0 + r + half * 8;
    xg[(size_t)row * kG + col] = acc[r] + bias;
  }
}

// ---------------------------------------------------------------------------
// Kernel 2: LSTM recurrence. grid = 2 (dir), block = 1024 (32 waves).
// Per step: gates[128,400] = hstate @ W_hh^T + xg[t]  (WMMA, 200 tiles),
// then gate activations update hstate/cstate and emit h_seq[t].
// ---------------------------------------------------------------------------
__global__ __launch_bounds__(1024) void k_lstm(
    const float* __restrict__ xg_f, const float* __restrict__ xg_b,
    const float* __restrict__ whh_f, const float* __restrict__ whh_b,
    float* __restrict__ hseq_f, float* __restrict__ hseq_b,
    float* __restrict__ gates_f, float* __restrict__ gates_b,
    float* __restrict__ hs_f, float* __restrict__ hs_b,
    float* __restrict__ cs_f, float* __restrict__ cs_b)
{
  const int dir = blockIdx.x;       // 0 = forward, 1 = backward
  const float* xg    = dir == 0 ? xg_f   : xg_b;
  const float* w_hh  = dir == 0 ? whh_f  : whh_b;
  float* hseq        = dir == 0 ? hseq_f : hseq_b;
  float* gates       = dir == 0 ? gates_f : gates_b;
  float* hstate      = dir == 0 ? hs_f   : hs_b;
  float* cstate      = dir == 0 ? cs_f   : cs_b;

  const int tid  = threadIdx.x;
  const int lane = tid & 31;
  const int wave = tid >> 5;
  const int half = lane >> 4;
  const int lq   = lane & 15;

  // zero-init state (h0 = c0 = 0)
  for (int i = tid; i < kB * kH; i += blockDim.x) { hstate[i] = 0.f; cstate[i] = 0.f; }
  __threadfence();
  __syncthreads();

  for (int s = 0; s < kT; ++s) {
    const int t = (dir == 0) ? s : (kT - 1 - s);
    const float* __restrict__ xgt = xg + (size_t)t * kB * kG;

    // ---- phase A: gates = hstate @ W_hh^T + xg[t] (8 x 25 = 200 tiles) ----
    for (int tile = wave; tile < (kB / 16) * kNT_G; tile += 32) {
      const int m0 = (tile / kNT_G) * 16;
      const int n0 = (tile % kNT_G) * 16;
      const float* __restrict__ arow = hstate + (size_t)(m0 + lq) * kH;
      const float* __restrict__ brow = w_hh + (size_t)(n0 + lq) * kH;

      v8f acc = {};
      for (int k0 = 0; k0 < kH; k0 += 4) {
        const int ka = k0 + half * 2;
        v2f av; av.x = arow[ka]; av.y = arow[ka + 1];
        v2f bv; bv.x = brow[ka]; bv.y = brow[ka + 1];
        acc = wmma_f32_4(av, bv, acc);
      }
      const int col = n0 + lq;
#pragma unroll
      for (int r = 0; r < 8; ++r) {
        const int row = m0 + r + half * 8;
        gates[(size_t)row * kG + col] = acc[r] + xgt[(size_t)row * kG + col];
      }
    }
    __threadfence();
    __syncthreads();

    // ---- phase B: activations, state update (PyTorch gate order i,f,g,o) ----
    for (int i = tid; i < kB * kH; i += blockDim.x) {
      const int b = i / kH;
      const int j = i % kH;
      const float* gb = gates + (size_t)b * kG;
      const float ig = sigmoidf(gb[j]);
      const float fg = sigmoidf(gb[kH + j]);
      const float gg = tanhf(gb[2 * kH + j]);
      const float og = sigmoidf(gb[3 * kH + j]);
      const float c  = fg * cstate[i] + ig * gg;
      const float h  = og * tanhf(c);
      cstate[i] = c;
      hstate[i] = h;
      hseq[(size_t)t * kB * kH + i] = h;
    }
    __threadfence();
    __syncthreads();
  }
}

// ---------------------------------------------------------------------------
// Kernel 3: emissions[t*B+b, l] = concat(h_f,h_b)[m,:200] . w_tag[l,:] + b_tag[l]
// N padded 25 -> 32 (2 N-tiles), masked store. One wave per tile.
// ---------------------------------------------------------------------------
__global__ __launch_bounds__(256) void k_emissions(
    const float* __restrict__ hf,     // [T*B, H]
    const float* __restrict__ hb,     // [T*B, H]
    const float* __restrict__ w_tag,  // [L, 2H]
    const float* __restrict__ b_tag,  // [L]
    float* __restrict__ em)           // [T*B, L]
{
  const int lane = threadIdx.x & 31;
  const int wave = threadIdx.x >> 5;
  const int gw   = blockIdx.x * (blockDim.x >> 5) + wave;
  if (gw >= kMT * 2) return;

  const int m0 = (gw >> 1) * 16;
  const int n0 = (gw & 1) * 16;
  const int half = lane >> 4;
  const int lq   = lane & 15;

  const int m = m0 + lq;
  const float* __restrict__ af = hf + (size_t)m * kH;
  const float* __restrict__ ab = hb + (size_t)m * kH;
  const int n = n0 + lq;
  const bool nvalid = (n < kL);
  const float* __restrict__ brow = w_tag + (size_t)(nvalid ? n : 0) * (2 * kH);

  v8f acc = {};
  for (int k0 = 0; k0 < 2 * kH; k0 += 4) {
    const int ka = k0 + half * 2;
    v2f av;
    av.x = (ka     < kH) ? af[ka]     : ab[ka - kH];
    av.y = (ka + 1 < kH) ? af[ka + 1] : ab[ka + 1 - kH];
    v2f bv;
    bv.x = nvalid ? brow[ka]     : 0.f;
    bv.y = nvalid ? brow[ka + 1] : 0.f;
    acc = wmma_f32_4(av, bv, acc);
  }

  const int col = n0 + lq;
  if (col < kL) {
    const float bias = b_tag[col];
#pragma unroll
    for (int r = 0; r < 8; ++r) {
      const int row = m0 + r + half * 8;
      em[(size_t)row * kL + col] = acc[r] + bias;
    }
  }
}

// ---------------------------------------------------------------------------
// Kernel 4: CRF NLL. Single block of 1024 threads.
// Gold path score (threads 0..127) + forward algorithm (all threads, ping-pong
// alpha in LDS) + final reduction to scalar.
// em layout: [T, B, L] (row m = t*B+b).
// ---------------------------------------------------------------------------
__global__ __launch_bounds__(1024) void k_crf(
    const float* __restrict__ em,       // [T,B,L]
    const int* __restrict__ tags,       // [B,T]
    const int* __restrict__ lengths,    // [B]
    const float* __restrict__ trans,    // [L,L]
    const float* __restrict__ start_t,  // [L]
    const float* __restrict__ end_t,    // [L]
    float* __restrict__ out)            // [1]
{
  __shared__ float sTrans[kL * kL];
  __shared__ float sAlpha[2][kB * kL];
  __shared__ float sGold[kB];
  __shared__ float sLogZ[kB];

  const int tid = threadIdx.x;
  for (int i = tid; i < kL * kL; i += blockDim.x) sTrans[i] = trans[i];
  __syncthreads();

  // ---- gold path score ----
  if (tid < kB) {
    const int b = tid;
    const int len = lengths[b];
    int tp = tags[b * kT];
    float score = start_t[tp] + em[b * kL + tp];
    for (int t = 1; t < kT; ++t) {
      const int tc = tags[b * kT + t];
      const float mk = (t < len) ? 1.f : 0.f;
      score += mk * (sTrans[tp * kL + tc] + em[(size_t)t * kB * kL + b * kL + tc]);
      tp = tc;
    }
    const int last = tags[b * kT + (len - 1)];
    sGold[b] = score + end_t[last];
  }

  // ---- forward algorithm (log partition) ----
  for (int i = tid; i < kB * kL; i += blockDim.x) {
    const int b = i / kL, l = i % kL;
    sAlpha[0][i] = start_t[l] + em[b * kL + l];
  }
  __syncthreads();

  int cur = 0;
  for (int t = 1; t < kT; ++t) {
    const float* __restrict__ emt = em + (size_t)t * kB * kL;
    for (int i = tid; i < kB * kL; i += blockDim.x) {
      const int b = i / kL, l2 = i % kL;
      const float* ar = &sAlpha[cur][b * kL];
      float mx = -INFINITY;
#pragma unroll
      for (int l = 0; l < kL; ++l)
        mx = fmaxf(mx, ar[l] + sTrans[l * kL + l2]);
      float sum = 0.f;
#pragma unroll
      for (int l = 0; l < kL; ++l)
        sum += expf(ar[l] + sTrans[l * kL + l2] - mx);
      const float nxt = mx + logf(sum) + emt[b * kL + l2];
      const bool valid = (t < lengths[b]);
      sAlpha[1 - cur][i] = valid ? nxt : ar[l2];
    }
    __syncthreads();
    cur = 1 - cur;
  }

  if (tid < kB) {
    const int b = tid;
    const float* ar = &sAlpha[cur][b * kL];
    float mx = -INFINITY;
#pragma unroll
    for (int l = 0; l < kL; ++l) mx = fmaxf(mx, ar[l] + end_t[l]);
    float sum = 0.f;
#pragma unroll
    for (int l = 0; l < kL; ++l) sum += expf(ar[l] + end_t[l] - mx);
    sLogZ[b] = mx + logf(sum);
  }
  __syncthreads();

  if (tid == 0) {
    float acc = 0.f;
    for (int b = 0; b < kB; ++b) acc += sLogZ[b] - sGold[b];
    out[0] = acc;
  }
}

// ---------------------------------------------------------------------------
// Host launcher
// ---------------------------------------------------------------------------
extern "C" void kernel_launch(void* const* d_in, const int* in_sizes, int n_in,
                              void* d_out, int out_size, void* d_ws, size_t ws_size,
                              hipStream_t stream) {
  const int*   tok      = (const int*)  d_in[0];
  const int*   tags     = (const int*)  d_in[1];
  const int*   lengths  = (const int*)  d_in[2];
  const float* word_emb = (const float*)d_in[3];
  const float* w_ih_f   = (const float*)d_in[4];
  const float* w_hh_f   = (const float*)d_in[5];
  const float* b_ih_f   = (const float*)d_in[6];
  const float* b_hh_f   = (const float*)d_in[7];
  const float* w_ih_b   = (const float*)d_in[8];
  const float* w_hh_b   = (const float*)d_in[9];
  const float* b_ih_b   = (const float*)d_in[10];
  const float* b_hh_b   = (const float*)d_in[11];
  const float* w_tag    = (const float*)d_in[12];
  const float* b_tag    = (const float*)d_in[13];
  const float* trans    = (const float*)d_in[14];
  const float* start_t  = (const float*)d_in[15];
  const float* end_t    = (const float*)d_in[16];
  float* out = (float*)d_out;

  // Workspace carve-up (floats). Total ~67.3M floats (~269 MB).
  float* w = (float*)d_ws;
  float* xg_f    = w;                       // [T*B, 400] = 26,214,400
  float* xg_b    = xg_f    + (size_t)kM * kG;
  float* hseq_f  = xg_b    + (size_t)kM * kG;   // [T*B, 100]
  float* hseq_b  = hseq_f  + (size_t)kM * kH;
  float* em      = hseq_b  + (size_t)kM * kH;   // [T*B, 25]
  float* gates_f = em      + (size_t)kM * kL;   // [128, 400]
  float* gates_b = gates_f + (size_t)kB * kG;
  float* hs_f    = gates_b + (size_t)kB * kG;   // [128, 100]
  float* cs_f    = hs_f    + (size_t)kB * kH;
  float* hs_b    = cs_f    + (size_t)kB * kH;
  float* cs_b    = hs_b    + (size_t)kB * kH;

  // 1) Input projections (embedding gather fused), one launch per direction.
  //    4096 M-tiles * 25 N-tiles = 102400 waves, 8 waves/block -> 12800 blocks.
  k_input_proj<<<12800, 256, 0, stream>>>(tok, word_emb, w_ih_f, b_ih_f, b_hh_f, xg_f);
  k_input_proj<<<12800, 256, 0, stream>>>(tok, word_emb, w_ih_b, b_ih_b, b_hh_b, xg_b);

  // 2) Sequential LSTM recurrences, one persistent block per direction.
  k_lstm<<<2, 1024, 0, stream>>>(xg_f, xg_b, w_hh_f, w_hh_b,
                                 hseq_f, hseq_b, gates_f, gates_b,
                                 hs_f, hs_b, cs_f, cs_b);

  // 3) hidden2tag emissions: 4096 M-tiles * 2 N-tiles = 8192 waves -> 1024 blocks.
  k_emissions<<<1024, 256, 0, stream>>>(hseq_f, hseq_b, w_tag, b_tag, em);

  // 4) CRF NLL -> scalar.
  k_crf<<<1, 1024, 0, stream>>>(em, tags, lengths, trans, start_t, end_t, out);
}